// CRFTagger_66907000537122
// MI455X (gfx1250) — compile-verified
//
#include <hip/hip_runtime.h>
#include <hip/hip_bf16.h>
#include <stdint.h>

// CRF Viterbi decode for B=1024, S=1024, T=20 (all-ones mask).
// Forward: 1 wave per batch, lane = tag, cross-lane max-plus via ds_bpermute,
//          emissions streamed into LDS with the CDNA5 Tensor Data Mover.
// Backward: register-resident backpointer chase via ds_bpermute, with the
//           next block's backpointer words prefetched past the serial chase.

#define CRF_B 1024
#define CRF_S 1024
#define CRF_T 20
#define CRF_SQ (CRF_S / 4)   // packed backpointer dwords per (batch, tag)
#define EM_CHUNK 64          // emission rows per LDS chunk (64*20*4 = 5120 B)
#define EM_NC (CRF_S / EM_CHUNK)

typedef unsigned int v4u  __attribute__((ext_vector_type(4)));
typedef int          v8i_ __attribute__((ext_vector_type(8)));
typedef int          v4i_ __attribute__((ext_vector_type(4)));

#if defined(__has_builtin)
#if __has_builtin(__builtin_amdgcn_tensor_load_to_lds) && \
    __has_builtin(__builtin_amdgcn_s_wait_tensorcnt)
#define CRF_HAVE_TDM 1
#endif
#endif

// amdgpu-toolchain (clang-23 / therock-10.0) uses the 6-arg builtin:
//   (uint32x4 g0, int32x8 g1, int32x4 g2, int32x4 g3, int32x8, i32 cpol)
// ROCm 7.2 (clang-22) uses the 5-arg form.
#if defined(__clang_major__) && __clang_major__ >= 23
#define CRF_TDM_6ARG 1
#endif

#ifdef CRF_HAVE_TDM
// 1-D contiguous TDM copy: n f32 elements from gsrc -> LDS byte offset ldsOff.
// D# per CDNA5 ISA:
//   group0: count=1, lds_addr, global_addr[56:0], type=2
//   group1: data_size=2 (4B), tensor_dim0=n, tensor_dim1=1, tile_dim0=n,
//           tile_dim1=1, strides=n; groups 2/3 zero (<=2D tensor).
__device__ __forceinline__ void tdm_issue_1d(const void* gsrc, unsigned ldsOff,
                                             int n) {
  unsigned long long ga = (unsigned long long)(uintptr_t)gsrc;
  v4u g0;
  g0.x = 1u;                                                  // count=1
  g0.y = ldsOff;                                              // lds_addr
  g0.z = (unsigned)ga;                                        // gaddr[31:0]
  g0.w = ((unsigned)(ga >> 32) & 0x01FFFFFFu) | (2u << 30);   // gaddr[56:32]|type=2
  v8i_ g1;
  g1[0] = (int)(2u << 16);                          // data_size = 4 bytes
  g1[1] = (int)(((unsigned)n & 0xFFFFu) << 16);     // tensor_dim0[15:0]
  g1[2] = (int)(((unsigned)n >> 16) | (1u << 16));  // tdim0[31:16] | tdim1=1
  g1[3] = (int)(((unsigned)n & 0xFFFFu) << 16);     // tdim1 hi=0 | tile_dim0=n
  g1[4] = 1;                                        // tile_dim1=1, tile_dim2=0
  g1[5] = n;                                        // tensor_dim0_stride lo32
  g1[6] = (int)(((unsigned)n & 0xFFFFu) << 16);     // stride0 hi=0 | stride1 lo16
  g1[7] = (int)((unsigned)n >> 16);                 // stride1[47:16]
  v4i_ z4 = {0, 0, 0, 0};
#ifdef CRF_TDM_6ARG
  v8i_ z8 = {0, 0, 0, 0, 0, 0, 0, 0};
  __builtin_amdgcn_tensor_load_to_lds(g0, g1, z4, z4, z8, 0);
#else
  __builtin_amdgcn_tensor_load_to_lds(g0, g1, z4, z4, 0);
#endif
}
#endif

// One Viterbi step for all 20 tags of one batch (lane = tag j).
// prev score broadcast across lanes with ds_bpermute; first-max argmax.
__device__ __forceinline__ void viterbi_step(float& score, float e,
                                             const float* t, int& arg) {
  float prev = score;
  float best = __shfl(prev, 0, 32) + t[0];
  arg = 0;
#pragma unroll
  for (int i = 1; i < CRF_T; ++i) {
    float ci = __shfl(prev, i, 32) + t[i];
    if (ci > best) { best = ci; arg = i; }  // strict > == first max (jnp.argmax)
  }
  score = best + e;
}

__global__ __launch_bounds__(32) void crf_forward(
    const float* __restrict__ em, const float* __restrict__ startT,
    const float* __restrict__ endT, const float* __restrict__ trans,
    unsigned* __restrict__ bp32, int* __restrict__ lastTag) {
  const int b = blockIdx.x;
  const int lane = threadIdx.x;
  const int jc = lane < CRF_T ? lane : CRF_T - 1;  // clamp idle lanes

  // transitions column jc in registers: t[i] = trans[i][jc]
  float t[CRF_T];
#pragma unroll
  for (int i = 0; i < CRF_T; ++i) t[i] = trans[i * CRF_T + jc];

  const float* emB = em + (size_t)b * CRF_S * CRF_T;
  unsigned* bpRow = bp32 + ((size_t)b * CRF_T + jc) * CRF_SQ;

  float score;
  unsigned pack = 0;
  int arg;

#ifdef CRF_HAVE_TDM
  __shared__ float ldsEm[2][EM_CHUNK * CRF_T];
  const unsigned ldsOff0 = (unsigned)(uintptr_t)&ldsEm[0][0];
  const unsigned ldsOff1 = (unsigned)(uintptr_t)&ldsEm[1][0];

  tdm_issue_1d(emB, ldsOff0, EM_CHUNK * CRF_T);
  __builtin_amdgcn_s_wait_tensorcnt(0);
  __asm__ volatile("" ::: "memory");

  score = startT[jc] + ldsEm[0][jc];  // s = 0

  // ---- chunk 0 (peeled: starts at s = 1, overlaps chunk-1 TDM load) ----
  {
    __asm__ volatile("" ::: "memory");
    tdm_issue_1d(emB + (size_t)EM_CHUNK * CRF_T, ldsOff1, EM_CHUNK * CRF_T);
    const float* buf = &ldsEm[0][0];
#pragma unroll 4
    for (int s = 1; s < EM_CHUNK; ++s) {
      float e = buf[s * CRF_T + jc];
      viterbi_step(score, e, t, arg);
      const int k = s - 1;
      pack |= (unsigned)arg << (8 * (k & 3));
      if ((k & 3) == 3) {
        if (lane < CRF_T) bpRow[k >> 2] = pack;
        pack = 0;
      }
    }
    __builtin_amdgcn_s_wait_tensorcnt(0);
    __asm__ volatile("" ::: "memory");
  }

  // ---- chunks 1 .. NC-1 ----
  for (int c = 1; c < EM_NC; ++c) {
    if (c + 1 < EM_NC) {
      __asm__ volatile("" ::: "memory");
      tdm_issue_1d(emB + (size_t)(c + 1) * EM_CHUNK * CRF_T,
                   (c & 1) ? ldsOff0 : ldsOff1, EM_CHUNK * CRF_T);
    }
    const float* buf = &ldsEm[c & 1][0];
    const int sBeg = c * EM_CHUNK;
#pragma unroll 4
    for (int so = 0; so < EM_CHUNK; ++so) {
      const int s = sBeg + so;
      float e = buf[so * CRF_T + jc];
      viterbi_step(score, e, t, arg);
      const int k = s - 1;
      pack |= (unsigned)arg << (8 * (k & 3));
      if ((k & 3) == 3) {
        if (lane < CRF_T) bpRow[k >> 2] = pack;
        pack = 0;
      }
    }
    if (c + 1 < EM_NC) {
      __builtin_amdgcn_s_wait_tensorcnt(0);
      __asm__ volatile("" ::: "memory");
    }
  }
#else
  score = startT[jc] + emB[jc];
#pragma unroll 4
  for (int s = 1; s < CRF_S; ++s) {
    float e = emB[(size_t)s * CRF_T + jc];
    viterbi_step(score, e, t, arg);
    const int k = s - 1;
    pack |= (unsigned)arg << (8 * (k & 3));
    if ((k & 3) == 3) {
      if (lane < CRF_T) bpRow[k >> 2] = pack;
      pack = 0;
    }
  }
#endif

  // tail: k = S-2 = 1022 ends mid-dword (bytes 0..2) -> flush
  if (((CRF_S - 2) & 3) != 3) {
    if (lane < CRF_T) bpRow[(CRF_S - 2) >> 2] = pack;
  }

  // last_tag = argmax_j(score + end[j]), first-index tie-break.
  float v = (lane < CRF_T) ? (score + endT[jc]) : -__builtin_huge_valf();
  int idx = lane;
#pragma unroll
  for (int off = 16; off >= 1; off >>= 1) {
    float ov = __shfl_xor(v, off, 32);
    int oi = __shfl_xor(idx, off, 32);
    if (ov > v || (ov == v && oi < idx)) { v = ov; idx = oi; }
  }
  if (lane == 0) lastTag[b] = idx;
}

__global__ __launch_bounds__(32) void crf_backward(
    const unsigned* __restrict__ bp32, const int* __restrict__ lastTag,
    int* __restrict__ out) {
  const int b = blockIdx.x;
  const int lane = threadIdx.x;
  const int jc = lane < CRF_T ? lane : CRF_T - 1;
  const unsigned* row = bp32 + ((size_t)b * CRF_T + jc) * CRF_SQ;
  int* outB = out + (size_t)b * CRF_S;

  int tag = lastTag[b];
  const int NB = CRF_S / 32;

  // lane j holds 32 steps (8 packed dwords) of its tag column in registers
  unsigned d[8], dn[8];
#pragma unroll
  for (int r = 0; r < 8; ++r) d[r] = row[(NB - 1) * 8 + r];

  for (int c = NB - 1; c >= 0; --c) {
    // Prefetch the next block's words; they overlap the serial chase below.
    if (c > 0) {
#pragma unroll
      for (int r = 0; r < 8; ++r) dn[r] = row[(c - 1) * 8 + r];
    }
    int my = 0;
#pragma unroll
    for (int pi = 31; pi >= 0; --pi) {
      if (!(pi == 31 && c == NB - 1)) {  // position S-1 seeds from lastTag
        unsigned w = (unsigned)__shfl((int)d[pi >> 2], tag, 32);
        tag = (int)((w >> (8 * (pi & 3))) & 0xFFu);
      }
      if (lane == pi) my = tag;
    }
    outB[c * 32 + lane] = my;  // coalesced 128B store per block of positions
    if (c > 0) {
#pragma unroll
      for (int r = 0; r < 8; ++r) d[r] = dn[r];
    }
  }
}

extern "C" void kernel_launch(void* const* d_in, const int* in_sizes, int n_in,
                              void* d_out, int out_size, void* d_ws,
                              size_t ws_size, hipStream_t stream) {
  (void)in_sizes; (void)n_in; (void)out_size; (void)ws_size;
  const float* em     = (const float*)d_in[0];  // [B,S,T]
  const float* startT = (const float*)d_in[1];  // [T]
  const float* endT   = (const float*)d_in[2];  // [T]
  const float* trans  = (const float*)d_in[3];  // [T,T]
  int* out = (int*)d_out;                       // [B,S] int32

  const size_t bpBytes = (size_t)CRF_B * CRF_T * CRF_SQ * sizeof(unsigned);
  unsigned* bp32 = (unsigned*)d_ws;                    // ~20 MB scratch
  int* lastTag = (int*)((char*)d_ws + bpBytes);        // [B]

  crf_forward<<<dim3(CRF_B), dim3(32), 0, stream>>>(em, startT, endT, trans,
                                                    bp32, lastTag);
  crf_backward<<<dim3(CRF_B), dim3(32), 0, stream>>>(bp32, lastTag, out);
}